// RcsFullStitchCrop_29910152249781
// MI455X (gfx1250) — compile-verified
//
#include <hip/hip_runtime.h>
#include <hip/hip_bf16.h>
#include <stdint.h>

// ---------------------------------------------------------------------------
// Scatter-copy of 256x256 patches:  out[perm[k]] = patch[k]  (inverse of the
// reference's gather by argsort(perm)).  Pure bandwidth problem: ~402 MB of
// traffic -> ~17 us floor at 23.3 TB/s.  Implemented with the CDNA5 async
// global<->LDS DMA path (ASYNCcnt), double-buffered, B128 per lane.
// ---------------------------------------------------------------------------

#define THREADS   256
#define JPT       8                       // float4 per thread per chunk
#define CHUNK_F4  (THREADS * JPT)         // 2048 float4 = 32 KB per buffer
#define TILE_F4   16384                   // 256*256 floats / 4 per (patch,plane)
#define NCHUNK    (TILE_F4 / CHUNK_F4)    // 8

__device__ __forceinline__ void async_load_b128(uint32_t lds_addr, uint32_t voff,
                                                uint64_t sbase) {
  // GVS: mem = SADDR + VADDR[31:0];   vdst = LDS byte address
  asm volatile("global_load_async_to_lds_b128 %0, %1, %2"
               :: "v"(lds_addr), "v"(voff), "s"(sbase) : "memory");
}

__device__ __forceinline__ void async_store_b128(uint32_t voff, uint32_t lds_addr,
                                                 uint64_t dbase) {
  asm volatile("global_store_async_from_lds_b128 %0, %1, %2"
               :: "v"(voff), "v"(lds_addr), "s"(dbase) : "memory");
}

__device__ __forceinline__ void wait_async0() {
  asm volatile("s_wait_asynccnt 0" ::: "memory");
}

__global__ __launch_bounds__(THREADS) void patch_scatter_async(
    const float* __restrict__ x, const float* __restrict__ y,
    const int* __restrict__ perm, float* __restrict__ out)
{
  __shared__ float4 stage[2 * CHUNK_F4];   // 64 KB double buffer

  const int p   = blockIdx.x;              // source patch 0..255
  const int pc  = blockIdx.y;              // 0,1 = x channels; 2 = y
  const int tid = threadIdx.x;
  const int ph  = p >> 4;                  // patch grid is 16x16
  const int pw  = p & 15;

  // destination slot: inverse permutation via scatter (perm[inv] = identity)
  const int d = __builtin_amdgcn_readfirstlane(perm[p]);

  uint64_t sbase, dbase;
  if (pc < 2) {
    sbase = (uint64_t)(x + (size_t)pc * 4096 * 4096
                         + (size_t)ph * 256 * 4096
                         + (size_t)pw * 256);
    dbase = (uint64_t)(out + (size_t)d * (2 * 256 * 256)
                           + (size_t)pc * (256 * 256));
  } else {
    sbase = (uint64_t)(y + (size_t)ph * 256 * 4096
                         + (size_t)pw * 256);
    dbase = (uint64_t)(out + (size_t)256 * 2 * 256 * 256   // after out_x section
                           + (size_t)d * (256 * 256));
  }

  // LDS byte address of this thread's j=0 slot in buffer 0
  const uint32_t lds0 = (uint32_t)(uint64_t)(uintptr_t)&stage[tid];

  // ---- prologue: load chunk 0 into buffer 0 -------------------------------
#pragma unroll
  for (int j = 0; j < JPT; ++j) {
    const int idx4 = j * THREADS + tid;        // float4 index within tile
    const int r = idx4 >> 6;                   // row (64 float4 per row)
    const int q = idx4 & 63;                   // float4 within row
    async_load_b128(lds0 + (uint32_t)(j * THREADS * 16),
                    (uint32_t)(r * 16384 + q * 16),   // src: row stride 16 KB
                    sbase);
  }

  // ---- steady state: drain, then issue next loads + current stores --------
  for (int k = 0; k < NCHUNK; ++k) {
    wait_async0();   // loads of chunk k done; stores of chunk k-1 done
    const uint32_t bufA = (uint32_t)(k & 1) * (CHUNK_F4 * 16);
    const uint32_t bufB = (uint32_t)((k + 1) & 1) * (CHUNK_F4 * 16);

    if (k + 1 < NCHUNK) {
#pragma unroll
      for (int j = 0; j < JPT; ++j) {
        const int idx4 = (k + 1) * CHUNK_F4 + j * THREADS + tid;
        const int r = idx4 >> 6;
        const int q = idx4 & 63;
        async_load_b128(lds0 + bufB + (uint32_t)(j * THREADS * 16),
                        (uint32_t)(r * 16384 + q * 16), sbase);
      }
    }
#pragma unroll
    for (int j = 0; j < JPT; ++j) {
      const int idx4 = k * CHUNK_F4 + j * THREADS + tid;
      const int r = idx4 >> 6;
      const int q = idx4 & 63;
      async_store_b128((uint32_t)(r * 1024 + q * 16),   // dst: row stride 1 KB
                       lds0 + bufA + (uint32_t)(j * THREADS * 16), dbase);
    }
  }
  // s_endpgm performs an implicit wait-idle, flushing the tail stores.
}

extern "C" void kernel_launch(void* const* d_in, const int* in_sizes, int n_in,
                              void* d_out, int out_size, void* d_ws, size_t ws_size,
                              hipStream_t stream) {
  (void)in_sizes; (void)n_in; (void)out_size; (void)d_ws; (void)ws_size;
  const float* x    = (const float*)d_in[0];   // [1,2,4096,4096] f32
  const float* y    = (const float*)d_in[1];   // [1,1,4096,4096] f32
  const int*   perm = (const int*)d_in[2];     // [256] i32 permutation
  float*       out  = (float*)d_out;           // px[inv] (33554432 f) ++ py[inv] (16777216 f)

  dim3 grid(256, 3);        // 256 patches x (2 x-channels + 1 y-channel)
  dim3 block(THREADS);
  patch_scatter_async<<<grid, block, 0, stream>>>(x, y, perm, out);
}